// GATLayer_23931557773463
// MI455X (gfx1250) — compile-verified
//
#include <hip/hip_runtime.h>
#include <hip/hip_bf16.h>

// ---------------------------------------------------------------------------
// GAT layer for MI455X (gfx1250).
//  N=50000 nodes, E=800000 edges, NODE_IN=128, EDGE_IN=64, EFT=32, OUT=64
// Dense GEMMs run on V_WMMA_F32_16X16X4_F32 (fp32 matrix pipe, wave32).
// Edge-level work is bandwidth-optimized: logits use precomputed per-node
// scalars p/q; softmax normalization is deferred into the final GEMM.
// ---------------------------------------------------------------------------

typedef __attribute__((ext_vector_type(2))) float v2f;
typedef __attribute__((ext_vector_type(8))) float v8f;

#define GAT_N 50000
#define GAT_E 800000

__device__ __forceinline__ v8f wmma_f32_k4(v2f a, v2f b, v8f c) {
  // 8 args: (neg_a, A, neg_b, B, c_mod, C, reuse_a, reuse_b)
  return __builtin_amdgcn_wmma_f32_16x16x4_f32(false, a, false, b, (short)0, c,
                                               false, false);
}

// order-preserving float <-> uint map for atomic segment-max
__device__ __forceinline__ unsigned f2ord(float f) {
  unsigned u = __float_as_uint(f);
  return (u & 0x80000000u) ? ~u : (u | 0x80000000u);
}
__device__ __forceinline__ float ord2f(unsigned u) {
  return (u & 0x80000000u) ? __uint_as_float(u & 0x7fffffffu)
                           : __uint_as_float(~u);
}

// ---------------------------------------------------------------------------
// zero scratch region (segmax | denom | agg). ordered-uint 0 == -inf.
// ---------------------------------------------------------------------------
__global__ void gat_zero_kernel(unsigned* __restrict__ p, int n) {
  int i = blockIdx.x * blockDim.x + threadIdx.x;
  if (i < n) p[i] = 0u;
}

// ---------------------------------------------------------------------------
// D[M,N] = A[M,K] @ W[N,K]^T    (plain epilogue)
// BM rows per block, BM/16 waves, LDS-staged tiles, fp32 WMMA 16x16x4.
// A staged with b128 loads (rows 16B aligned; SA=K+4 keeps LDS rows aligned
// AND bank-conflict-free: bank step 4 per row -> 16 distinct banks).
// ---------------------------------------------------------------------------
template <int K, int N, int BM>
__global__ __launch_bounds__((BM / 16) * 32) void gat_gemm_wt(
    const float* __restrict__ A, const float* __restrict__ W,
    float* __restrict__ D, int M) {
  constexpr int NT = (BM / 16) * 32;
  constexpr int SA = K + 4;
  __shared__ float Alds[BM * SA];
  __shared__ float Blds[K * N];

  const int tid = threadIdx.x;
  const int mbase = blockIdx.x * BM;
  const bool full = (mbase + BM) <= M;  // block-uniform fast path

  // stage A tile with float4 loads (row-clamped for the ragged last block)
  for (int idx = tid; idx < BM * (K / 4); idx += NT) {
    int r = idx / (K / 4), c4 = idx % (K / 4);
    int row = mbase + r;
    if (row >= M) row = M - 1;
    float4 v = ((const float4*)(A + (long)row * K))[c4];
    *(float4*)&Alds[r * SA + c4 * 4] = v;
  }
  // stage B = W^T  (B[k][n] = W[n*K + k])
  for (int idx = tid; idx < K * N; idx += NT) {
    int k = idx / N, n = idx % N;
    Blds[idx] = W[(long)n * K + k];
  }
  __syncthreads();

  const int w = tid >> 5, lane = tid & 31;
  const int mrow = lane & 15;      // A row / B col within 16
  const int kk = (lane >> 4) * 2;  // lane-half K offset

  for (int c0 = 0; c0 < N; c0 += 16) {
    v8f acc = {};
#pragma unroll 4
    for (int k0 = 0; k0 < K; k0 += 4) {
      v2f a, b;
      const float* ap = &Alds[(w * 16 + mrow) * SA + k0 + kk];
      a.x = ap[0];
      a.y = ap[1];
      b.x = Blds[(k0 + kk) * N + c0 + mrow];
      b.y = Blds[(k0 + kk + 1) * N + c0 + mrow];
      acc = wmma_f32_k4(a, b, acc);
    }
    const int rbase = mbase + w * 16 + (lane >> 4) * 8;
    const int col = c0 + (lane & 15);
    if (full) {
#pragma unroll
      for (int r = 0; r < 8; ++r) D[(long)(rbase + r) * N + col] = acc[r];
    } else {
#pragma unroll
      for (int r = 0; r < 8; ++r) {
        int row = rbase + r;
        if (row < M) D[(long)row * N + col] = acc[r];
      }
    }
  }
}

// ---------------------------------------------------------------------------
// p[n] = z[n]·Wa[0:64], q[n] = z[n]·Wa[96:160]   — one wave per node
// ---------------------------------------------------------------------------
__global__ void gat_pq_kernel(const float* __restrict__ z,
                              const float* __restrict__ Wa,
                              float* __restrict__ p, float* __restrict__ q,
                              int N) {
  int wid = (blockIdx.x * blockDim.x + threadIdx.x) >> 5;
  int lane = threadIdx.x & 31;
  if (wid >= N) return;
  float z1 = z[(long)wid * 64 + lane];
  float z2 = z[(long)wid * 64 + 32 + lane];
  float pp = z1 * Wa[lane] + z2 * Wa[32 + lane];
  float qq = z1 * Wa[96 + lane] + z2 * Wa[128 + lane];
#pragma unroll
  for (int off = 16; off; off >>= 1) {
    pp += __shfl_xor(pp, off, 32);
    qq += __shfl_xor(qq, off, 32);
  }
  if (lane == 0) {
    p[wid] = pp;
    q[wid] = qq;
  }
}

// ---------------------------------------------------------------------------
// per-edge logit s = leaky_relu(p[src] + ef·Wa[64:96] + q[dst]);
// segment-max via ordered-uint atomicMax.  one wave per edge.
// ---------------------------------------------------------------------------
__global__ void gat_logit_kernel(const float* __restrict__ ef,
                                 const int* __restrict__ src,
                                 const int* __restrict__ dst,
                                 const float* __restrict__ p,
                                 const float* __restrict__ q,
                                 const float* __restrict__ Wa,
                                 float* __restrict__ sv,
                                 unsigned* __restrict__ segmax, int E) {
  int e = (blockIdx.x * blockDim.x + threadIdx.x) >> 5;
  int lane = threadIdx.x & 31;
  if (e >= E) return;
  float r = ef[(long)e * 32 + lane] * Wa[64 + lane];
#pragma unroll
  for (int off = 16; off; off >>= 1) r += __shfl_xor(r, off, 32);
  if (lane == 0) {
    float a = p[src[e]] + r + q[dst[e]];
    float s = a > 0.0f ? a : 0.01f * a;  // leaky_relu(0.01)
    sv[e] = s;
    atomicMax(&segmax[dst[e]], f2ord(s));
  }
}

// ---------------------------------------------------------------------------
// ex = exp(s - segmax[dst]); denom[dst] += ex.   one thread per edge.
// ---------------------------------------------------------------------------
__global__ void gat_exp_kernel(const int* __restrict__ dst,
                               const unsigned* __restrict__ segmax,
                               float* __restrict__ sv,
                               float* __restrict__ denom, int E) {
  int e = blockIdx.x * blockDim.x + threadIdx.x;
  if (e >= E) return;
  int d = dst[e];
  float ex = expf(sv[e] - ord2f(segmax[d]));
  sv[e] = ex;
  atomicAdd(&denom[d], ex);
}

// ---------------------------------------------------------------------------
// agg[dst] += ex * [z[src](64) | ef(32)]   — one wave per edge, 3 lanes-wide
// (normalization by denom is deferred to the output GEMM)
// ---------------------------------------------------------------------------
__global__ void gat_agg_kernel(const float* __restrict__ z,
                               const float* __restrict__ ef,
                               const int* __restrict__ src,
                               const int* __restrict__ dst,
                               const float* __restrict__ sv,
                               float* __restrict__ agg, int E) {
  int e = (blockIdx.x * blockDim.x + threadIdx.x) >> 5;
  int lane = threadIdx.x & 31;
  if (e >= E) return;
  int s = src[e], d = dst[e];
  float ex = sv[e];
  float v0 = z[(long)s * 64 + lane];
  float v1 = z[(long)s * 64 + 32 + lane];
  float v2 = ef[(long)e * 32 + lane];
  atomicAdd(&agg[(long)d * 96 + lane], ex * v0);
  atomicAdd(&agg[(long)d * 96 + 32 + lane], ex * v1);
  atomicAdd(&agg[(long)d * 96 + 64 + lane], ex * v2);
}

// ---------------------------------------------------------------------------
// out = ((agg/denom) @ W2^T) where denom>0 else z.   WMMA f32, K=96, N=64.
// ---------------------------------------------------------------------------
template <int K, int N, int BM>
__global__ __launch_bounds__((BM / 16) * 32) void gat_out_gemm(
    const float* __restrict__ agg, const float* __restrict__ denom,
    const float* __restrict__ W2, const float* __restrict__ z,
    float* __restrict__ out, int M) {
  constexpr int NT = (BM / 16) * 32;
  constexpr int SA = K + 4;
  __shared__ float Alds[BM * SA];
  __shared__ float Blds[K * N];
  __shared__ float scs[BM];

  const int tid = threadIdx.x;
  const int mbase = blockIdx.x * BM;
  const bool full = (mbase + BM) <= M;

  if (tid < BM) {
    int row = mbase + tid;
    float dn = (row < M) ? denom[row] : 0.0f;
    scs[tid] = dn > 0.0f ? 1.0f / dn : 0.0f;
  }
  __syncthreads();

  for (int idx = tid; idx < BM * (K / 4); idx += NT) {
    int r = idx / (K / 4), c4 = idx % (K / 4);
    int row = mbase + r;
    if (row >= M) row = M - 1;
    float4 v = ((const float4*)(agg + (long)row * K))[c4];
    float sc = scs[r];
    v.x *= sc; v.y *= sc; v.z *= sc; v.w *= sc;
    *(float4*)&Alds[r * SA + c4 * 4] = v;
  }
  for (int idx = tid; idx < K * N; idx += NT) {
    int k = idx / N, n = idx % N;
    Blds[idx] = W2[(long)n * K + k];
  }
  __syncthreads();

  const int w = tid >> 5, lane = tid & 31;
  const int mrow = lane & 15;
  const int kk = (lane >> 4) * 2;

  for (int c0 = 0; c0 < N; c0 += 16) {
    v8f acc = {};
#pragma unroll 4
    for (int k0 = 0; k0 < K; k0 += 4) {
      v2f a, b;
      const float* ap = &Alds[(w * 16 + mrow) * SA + k0 + kk];
      a.x = ap[0];
      a.y = ap[1];
      b.x = Blds[(k0 + kk) * N + c0 + mrow];
      b.y = Blds[(k0 + kk + 1) * N + c0 + mrow];
      acc = wmma_f32_k4(a, b, acc);
    }
    const int rbase = mbase + w * 16 + (lane >> 4) * 8;
    const int col = c0 + (lane & 15);
    if (full) {
#pragma unroll
      for (int r = 0; r < 8; ++r) {
        int row = rbase + r;
        bool has_edges = denom[row] > 0.0f;
        out[(long)row * N + col] = has_edges ? acc[r] : z[(long)row * N + col];
      }
    } else {
#pragma unroll
      for (int r = 0; r < 8; ++r) {
        int row = rbase + r;
        if (row < M) {
          bool has_edges = denom[row] > 0.0f;
          out[(long)row * N + col] =
              has_edges ? acc[r] : z[(long)row * N + col];
        }
      }
    }
  }
}

// ---------------------------------------------------------------------------
// host launcher
// ---------------------------------------------------------------------------
extern "C" void kernel_launch(void* const* d_in, const int* in_sizes, int n_in,
                              void* d_out, int out_size, void* d_ws,
                              size_t ws_size, hipStream_t stream) {
  (void)in_sizes; (void)n_in; (void)out_size; (void)ws_size;
  const int N = GAT_N, E = GAT_E;

  const float* h   = (const float*)d_in[0];  // [N,128]
  const float* efi = (const float*)d_in[1];  // [E,64]
  const int* src   = (const int*)d_in[2];    // [E]
  const int* dst   = (const int*)d_in[3];    // [E]
  const float* Wn  = (const float*)d_in[4];  // [64,128]
  const float* We  = (const float*)d_in[5];  // [32,64]
  const float* Wa  = (const float*)d_in[6];  // [1,160]
  const float* W2  = (const float*)d_in[7];  // [64,96]
  float* out = (float*)d_out;                // [N,64]
  float* ws = (float*)d_ws;

  // workspace layout (floats): ~138.4 MB total; all region starts 16B aligned
  size_t off_z = 0;                    // N*64
  size_t off_p = off_z + (size_t)N * 64;
  size_t off_q = off_p + N;
  size_t off_ef = off_q + N;           // E*32
  size_t off_sv = off_ef + (size_t)E * 32;
  size_t off_segmax = off_sv + E;      // N   (zeroed; ordered-uint -inf == 0)
  size_t off_denom = off_segmax + N;   // N   (zeroed)
  size_t off_agg = off_denom + N;      // N*96 (zeroed)

  float* z = ws + off_z;
  float* p = ws + off_p;
  float* q = ws + off_q;
  float* ef = ws + off_ef;
  float* sv = ws + off_sv;
  unsigned* segmax = (unsigned*)(ws + off_segmax);
  float* denom = ws + off_denom;
  float* agg = ws + off_agg;

  // 1. zero segmax|denom|agg (contiguous N*98 words)
  {
    int n = N * 98;
    gat_zero_kernel<<<(n + 255) / 256, 256, 0, stream>>>(segmax, n);
  }
  // 2. z = h @ Wn^T   (M=N, K=128, Nc=64, 32 rows/block -> 64 threads)
  gat_gemm_wt<128, 64, 32><<<(N + 31) / 32, 64, 0, stream>>>(h, Wn, z, N);
  // 3. per-node p,q scalars (wave per node)
  gat_pq_kernel<<<(N * 32 + 255) / 256, 256, 0, stream>>>(z, Wa, p, q, N);
  // 4. ef = edge_feat @ We^T  (M=E, K=64, Nc=32, 64 rows/block -> 128 threads)
  gat_gemm_wt<64, 32, 64><<<(E + 63) / 64, 128, 0, stream>>>(efi, We, ef, E);
  // 5. edge logits + segment max (wave per edge)
  gat_logit_kernel<<<(E * 32 + 255) / 256, 256, 0, stream>>>(
      ef, src, dst, p, q, Wa, sv, segmax, E);
  // 6. exp + denom (thread per edge)
  gat_exp_kernel<<<(E + 255) / 256, 256, 0, stream>>>(dst, segmax, sv, denom,
                                                      E);
  // 7. weighted scatter aggregation (wave per edge)
  gat_agg_kernel<<<(E * 32 + 255) / 256, 256, 0, stream>>>(z, ef, src, dst, sv,
                                                           agg, E);
  // 8. out = select(deg>0, (agg/denom)@W2^T, z)
  gat_out_gemm<96, 64, 64><<<(N + 63) / 64, 128, 0, stream>>>(agg, denom, W2,
                                                              z, out, N);
}